// I2I_OptimalTransportAligner_8340826489177
// MI455X (gfx1250) — compile-verified
//
#include <hip/hip_runtime.h>
#include <hip/hip_bf16.h>

typedef __attribute__((ext_vector_type(2))) float v2f;
typedef __attribute__((ext_vector_type(8))) float v8f;

#define NCOL 9216      // columns per feature matrix (96*96)
#define LDIM 512       // feature length (4*128)
#define NM   ((size_t)NCOL * (size_t)NCOL)

// ---------------------------------------------------------------------------
// Phase 1a: column squared norms of both feature matrices.
// grid 72 x 256: blocks 0..35 -> feats1, 36..71 -> feats2.
__global__ void colnorms_kernel(const float* __restrict__ A,
                                const float* __restrict__ B,
                                float* __restrict__ a2, float* __restrict__ b2) {
    int which = blockIdx.x / 36;
    int c = (blockIdx.x % 36) * 256 + threadIdx.x;
    const float* src = which ? B : A;
    float s = 0.0f;
    for (int l = 0; l < LDIM; ++l) {
        float x = src[(size_t)l * NCOL + c];
        s += x * x;
    }
    (which ? b2 : a2)[c] = s;
}

// ---------------------------------------------------------------------------
// Phase 1b: C[n,m] = sqrt(max(a2[n] + b2[m] - 2 * sum_l A[l,n]*B[l,m], 0))
// fp32 WMMA 16x16x4.  128 threads = 4 waves, each wave owns a 32x32 tile,
// block covers 64x64.  grid (144,144).
__device__ __forceinline__ void dist_epilogue(float* __restrict__ C, const v8f& g,
                                              int nb, int mb, int half, int lo,
                                              const float* __restrict__ a2,
                                              const float* __restrict__ b2) {
    int m = mb + lo;
    float bm = b2[m];
#pragma unroll
    for (int r = 0; r < 8; ++r) {
        int n = nb + r + 8 * half;
        float d2 = a2[n] + bm - 2.0f * g[r];
        C[(size_t)n * NCOL + m] = sqrtf(fmaxf(d2, 0.0f));
    }
}

__global__ void __launch_bounds__(128)
gram_dist_kernel(const float* __restrict__ A, const float* __restrict__ B,
                 const float* __restrict__ a2, const float* __restrict__ b2,
                 float* __restrict__ C) {
    int t = threadIdx.x & 31;
    int w = threadIdx.x >> 5;
    int lo = t & 15, half = t >> 4;
    int n0 = blockIdx.y * 64 + (w >> 1) * 32;
    int m0 = blockIdx.x * 64 + (w & 1) * 32;

    v8f acc00 = {}, acc01 = {}, acc10 = {}, acc11 = {};
    for (int k = 0; k < LDIM; k += 4) {
        int ka = k + 2 * half;
        const float* Ar0 = A + (size_t)ka * NCOL;
        const float* Ar1 = A + (size_t)(ka + 1) * NCOL;
        const float* Br0 = B + (size_t)ka * NCOL;
        const float* Br1 = B + (size_t)(ka + 1) * NCOL;
        v2f a0, a1, b0, b1;
        a0.x = Ar0[n0 + lo];      a0.y = Ar1[n0 + lo];
        a1.x = Ar0[n0 + 16 + lo]; a1.y = Ar1[n0 + 16 + lo];
        b0.x = Br0[m0 + lo];      b0.y = Br1[m0 + lo];
        b1.x = Br0[m0 + 16 + lo]; b1.y = Br1[m0 + 16 + lo];
        acc00 = __builtin_amdgcn_wmma_f32_16x16x4_f32(false, a0, false, b0, (short)0, acc00, false, false);
        acc01 = __builtin_amdgcn_wmma_f32_16x16x4_f32(false, a0, false, b1, (short)0, acc01, false, false);
        acc10 = __builtin_amdgcn_wmma_f32_16x16x4_f32(false, a1, false, b0, (short)0, acc10, false, false);
        acc11 = __builtin_amdgcn_wmma_f32_16x16x4_f32(false, a1, false, b1, (short)0, acc11, false, false);
    }
    dist_epilogue(C, acc00, n0,      m0,      half, lo, a2, b2);
    dist_epilogue(C, acc01, n0,      m0 + 16, half, lo, a2, b2);
    dist_epilogue(C, acc10, n0 + 16, m0,      half, lo, a2, b2);
    dist_epilogue(C, acc11, n0 + 16, m0 + 16, half, lo, a2, b2);
}

// ---------------------------------------------------------------------------
// Phase 2: Sinkhorn.  K = exp(-10*C) recomputed in-register (bandwidth-bound).
__global__ void init_uv_kernel(float* u, float* v, int* done) {
    int i = blockIdx.x * 256 + threadIdx.x;
    if (i < NCOL) { u[i] = 1.0f; v[i] = 1.0f; }
    if (i == 0) *done = 0;
}

// one block per row: u[i] = 1/(sum_j exp(-10*C[i,j])*v[j] + 100)
__global__ void sinkhorn_row_kernel(const float* __restrict__ C,
                                    const float* __restrict__ v,
                                    float* __restrict__ u, float* __restrict__ d2u,
                                    const int* __restrict__ done) {
    if (*done) return;
    __shared__ float red[256];
    int i = blockIdx.x;
    const float* row = C + (size_t)i * NCOL;
    float s = 0.0f;
    for (int j = threadIdx.x; j < NCOL; j += 256)
        s += __expf(-10.0f * row[j]) * v[j];
    red[threadIdx.x] = s;
    __syncthreads();
    for (int off = 128; off > 0; off >>= 1) {
        if (threadIdx.x < off) red[threadIdx.x] += red[threadIdx.x + off];
        __syncthreads();
    }
    if (threadIdx.x == 0) {
        float nu = 1.0f / (red[0] + 100.0f);
        float d = nu - u[i];
        d2u[i] = d * d;
        u[i] = nu;
    }
}

// column partials: grid (36, 36); block (x,y) sums rows [y*256, y*256+256)
__global__ void sinkhorn_colpart_kernel(const float* __restrict__ C,
                                        const float* __restrict__ u,
                                        float* __restrict__ tpart,
                                        const int* __restrict__ done) {
    if (*done) return;
    int j = blockIdx.x * 256 + threadIdx.x;
    int i0 = blockIdx.y * 256;
    float s = 0.0f;
    for (int i = i0; i < i0 + 256; ++i)
        s += __expf(-10.0f * C[(size_t)i * NCOL + j]) * u[i];
    tpart[(size_t)blockIdx.y * NCOL + j] = s;
}

__global__ void sinkhorn_colfin_kernel(const float* __restrict__ tpart,
                                       float* __restrict__ v, float* __restrict__ d2v,
                                       const int* __restrict__ done) {
    if (*done) return;
    int j = blockIdx.x * 256 + threadIdx.x;
    float s = 0.0f;
    for (int c = 0; c < 36; ++c) s += tpart[(size_t)c * NCOL + j];
    float nv = 1.0f / (s + 100.0f);
    float d = nv - v[j];
    d2v[j] = d * d;
    v[j] = nv;
}

// deterministic convergence check: ||du||^2 < 1e-4 && ||dv||^2 < 1e-4
__global__ void sinkhorn_update_kernel(const float* __restrict__ d2u,
                                       const float* __restrict__ d2v,
                                       int* done) {
    __shared__ float ru[256], rv[256];
    float su = 0.0f, sv = 0.0f;
    for (int i = threadIdx.x; i < NCOL; i += 256) { su += d2u[i]; sv += d2v[i]; }
    ru[threadIdx.x] = su; rv[threadIdx.x] = sv;
    __syncthreads();
    for (int off = 128; off > 0; off >>= 1) {
        if (threadIdx.x < off) { ru[threadIdx.x] += ru[threadIdx.x + off];
                                 rv[threadIdx.x] += rv[threadIdx.x + off]; }
        __syncthreads();
    }
    if (threadIdx.x == 0) {
        if (ru[0] < 1e-4f && rv[0] < 1e-4f) *done = 1;
    }
}

// ---------------------------------------------------------------------------
// Phase 3a: in-place T[n,m] = exp(-0.1*C[n,m]) * u[n] * v[m]   (C overwritten)
// grid (36, 9216)
__global__ void make_T_kernel(float* __restrict__ CT,
                              const float* __restrict__ u,
                              const float* __restrict__ v) {
    int n = blockIdx.y;
    int m = blockIdx.x * 256 + threadIdx.x;
    size_t idx = (size_t)n * NCOL + m;
    CT[idx] = __expf(-0.1f * CT[idx]) * u[n] * v[m];
}

// ---------------------------------------------------------------------------
// Phase 3b: a1[l,n] = sum_m Bfeats[l,m] * T[n,m]   (B @ T^T)
// grid (144, 8), 128 threads.  B-op fragment is two contiguous floats of a
// T row -> v2f load.
__global__ void __launch_bounds__(128)
gemm_alignedA_kernel(const float* __restrict__ Bf, const float* __restrict__ T,
                     float* __restrict__ out) {
    int t = threadIdx.x & 31;
    int w = threadIdx.x >> 5;
    int lo = t & 15, half = t >> 4;
    int l0 = blockIdx.y * 64 + (w >> 1) * 32;
    int n0 = blockIdx.x * 64 + (w & 1) * 32;

    v8f acc00 = {}, acc01 = {}, acc10 = {}, acc11 = {};
    for (int k = 0; k < NCOL; k += 4) {
        int kk = k + 2 * half;
        v2f a0 = *(const v2f*)(Bf + (size_t)(l0 + lo) * NCOL + kk);
        v2f a1 = *(const v2f*)(Bf + (size_t)(l0 + 16 + lo) * NCOL + kk);
        v2f b0 = *(const v2f*)(T + (size_t)(n0 + lo) * NCOL + kk);
        v2f b1 = *(const v2f*)(T + (size_t)(n0 + 16 + lo) * NCOL + kk);
        acc00 = __builtin_amdgcn_wmma_f32_16x16x4_f32(false, a0, false, b0, (short)0, acc00, false, false);
        acc01 = __builtin_amdgcn_wmma_f32_16x16x4_f32(false, a0, false, b1, (short)0, acc01, false, false);
        acc10 = __builtin_amdgcn_wmma_f32_16x16x4_f32(false, a1, false, b0, (short)0, acc10, false, false);
        acc11 = __builtin_amdgcn_wmma_f32_16x16x4_f32(false, a1, false, b1, (short)0, acc11, false, false);
    }
#pragma unroll
    for (int r = 0; r < 8; ++r) {
        out[(size_t)(l0 + r + 8 * half) * NCOL + n0 + lo]           = acc00[r];
        out[(size_t)(l0 + r + 8 * half) * NCOL + n0 + 16 + lo]      = acc01[r];
        out[(size_t)(l0 + 16 + r + 8 * half) * NCOL + n0 + lo]      = acc10[r];
        out[(size_t)(l0 + 16 + r + 8 * half) * NCOL + n0 + 16 + lo] = acc11[r];
    }
}

// Phase 3c: a2[l,m] = sum_n Afeats[l,n] * T[n,m]   (A @ T)
__global__ void __launch_bounds__(128)
gemm_alignedB_kernel(const float* __restrict__ Af, const float* __restrict__ T,
                     float* __restrict__ out) {
    int t = threadIdx.x & 31;
    int w = threadIdx.x >> 5;
    int lo = t & 15, half = t >> 4;
    int l0 = blockIdx.y * 64 + (w >> 1) * 32;
    int m0 = blockIdx.x * 64 + (w & 1) * 32;

    v8f acc00 = {}, acc01 = {}, acc10 = {}, acc11 = {};
    for (int k = 0; k < NCOL; k += 4) {
        int kk = k + 2 * half;
        v2f a0 = *(const v2f*)(Af + (size_t)(l0 + lo) * NCOL + kk);
        v2f a1 = *(const v2f*)(Af + (size_t)(l0 + 16 + lo) * NCOL + kk);
        const float* Tr0 = T + (size_t)kk * NCOL;
        const float* Tr1 = T + (size_t)(kk + 1) * NCOL;
        v2f b0, b1;
        b0.x = Tr0[m0 + lo];      b0.y = Tr1[m0 + lo];
        b1.x = Tr0[m0 + 16 + lo]; b1.y = Tr1[m0 + 16 + lo];
        acc00 = __builtin_amdgcn_wmma_f32_16x16x4_f32(false, a0, false, b0, (short)0, acc00, false, false);
        acc01 = __builtin_amdgcn_wmma_f32_16x16x4_f32(false, a0, false, b1, (short)0, acc01, false, false);
        acc10 = __builtin_amdgcn_wmma_f32_16x16x4_f32(false, a1, false, b0, (short)0, acc10, false, false);
        acc11 = __builtin_amdgcn_wmma_f32_16x16x4_f32(false, a1, false, b1, (short)0, acc11, false, false);
    }
#pragma unroll
    for (int r = 0; r < 8; ++r) {
        out[(size_t)(l0 + r + 8 * half) * NCOL + m0 + lo]           = acc00[r];
        out[(size_t)(l0 + r + 8 * half) * NCOL + m0 + 16 + lo]      = acc01[r];
        out[(size_t)(l0 + 16 + r + 8 * half) * NCOL + m0 + lo]      = acc10[r];
        out[(size_t)(l0 + 16 + r + 8 * half) * NCOL + m0 + 16 + lo] = acc11[r];
    }
}

// ---------------------------------------------------------------------------
extern "C" void kernel_launch(void* const* d_in, const int* in_sizes, int n_in,
                              void* d_out, int out_size, void* d_ws, size_t ws_size,
                              hipStream_t stream) {
    (void)in_sizes; (void)n_in; (void)out_size; (void)ws_size;
    const float* img1 = (const float*)d_in[0];   // feats1: [512, 9216]
    const float* img2 = (const float*)d_in[1];   // feats2: [512, 9216]
    float* out = (float*)d_out;                  // [a1 | a2], each 512*9216

    char* ws = (char*)d_ws;
    float* CT    = (float*)ws;                        // 9216*9216 cost matrix -> later T
    float* u     = (float*)(ws + NM * sizeof(float));
    float* v     = u + NCOL;
    float* d2u   = v + NCOL;
    float* d2v   = d2u + NCOL;
    float* a2    = d2v + NCOL;
    float* b2    = a2 + NCOL;
    float* tpart = b2 + NCOL;                         // 36 * 9216 partials
    int*   done  = (int*)(tpart + 36 * NCOL);

    // Phase 1: cost matrix C via fp32 WMMA gram + fused distance epilogue
    colnorms_kernel<<<72, 256, 0, stream>>>(img1, img2, a2, b2);
    gram_dist_kernel<<<dim3(144, 144), 128, 0, stream>>>(img1, img2, a2, b2, CT);

    // Phase 2: Sinkhorn with early-stop freeze (reference semantics)
    init_uv_kernel<<<36, 256, 0, stream>>>(u, v, done);
    for (int it = 0; it < 50; ++it) {
        sinkhorn_row_kernel<<<NCOL, 256, 0, stream>>>(CT, v, u, d2u, done);
        sinkhorn_colpart_kernel<<<dim3(36, 36), 256, 0, stream>>>(CT, u, tpart, done);
        sinkhorn_colfin_kernel<<<36, 256, 0, stream>>>(tpart, v, d2v, done);
        sinkhorn_update_kernel<<<1, 256, 0, stream>>>(d2u, d2v, done);
    }

    // Phase 3: T in place, then the two output GEMMs
    make_T_kernel<<<dim3(36, NCOL), 256, 0, stream>>>(CT, u, v);
    gemm_alignedA_kernel<<<dim3(144, 8), 128, 0, stream>>>(img2, CT, out);
    gemm_alignedB_kernel<<<dim3(144, 8), 128, 0, stream>>>(img1, CT, out + (size_t)LDIM * NCOL);
}